// DiscriminativeLoss_13692355740436
// MI455X (gfx1250) — compile-verified
//
#include <hip/hip_runtime.h>
#include <hip/hip_bf16.h>
#include <math.h>

// Problem constants (match the reference)
#define B_DIM 8
#define D_DIM 32
#define N_DIM 65536
#define NSEG  21          // labels 0..20 (0 = ignore)
#define NLAB  20
#define DELTA_V 0.5f
#define DELTA_D 3.0f
#define CHUNKS 128        // blocks per batch for streaming passes

// Workspace layout (in floats)
#define OFF_COUNTS 0                       // B*21
#define OFF_SUMS   192                     // B*21*32 = 5376
#define OFF_VSUM   (192 + 5376)            // B*21
#define OFF_VAR    (OFF_VSUM + 192)        // B
#define OFF_DIST   (OFF_VAR + 8)           // B
#define WS_FLOATS  (OFF_DIST + 8 + 16)

typedef float v2f __attribute__((ext_vector_type(2)));
typedef float v8f __attribute__((ext_vector_type(8)));

__device__ __forceinline__ float wave_reduce_add(float v) {
    #pragma unroll
    for (int m = 16; m >= 1; m >>= 1) v += __shfl_xor(v, m, 32);
    return v;
}

// ---------------------------------------------------------------- kernel 0
__global__ void zero_ws_kernel(float* ws, float* out) {
    int idx = blockIdx.x * blockDim.x + threadIdx.x;
    if (idx < WS_FLOATS) ws[idx] = 0.0f;
    if (idx < 3) out[idx] = 0.0f;   // out[2] = reg_loss stays 0
}

// ---------------------------------------------------------------- kernel 1
// counts[b][s] and sums[b][s][d] via LDS accumulation + global atomics.
__global__ void accum_kernel(const float* __restrict__ emb,
                             const int* __restrict__ seg,
                             float* __restrict__ counts_g,
                             float* __restrict__ sums_g) {
    __shared__ float s_counts[NSEG];
    __shared__ float s_sums[NSEG * D_DIM];
    const int b   = blockIdx.y;
    const int tid = threadIdx.x;

    for (int i = tid; i < NSEG; i += blockDim.x) s_counts[i] = 0.0f;
    for (int i = tid; i < NSEG * D_DIM; i += blockDim.x) s_sums[i] = 0.0f;
    __syncthreads();

    const int pix = N_DIM / CHUNKS;
    const int n0  = blockIdx.x * pix;
    const size_t ebase = (size_t)b * D_DIM * N_DIM;

    for (int n = n0 + tid; n < n0 + pix; n += blockDim.x) {
        const int s = seg[(size_t)b * N_DIM + n];
        atomicAdd(&s_counts[s], 1.0f);
        const float* ep = emb + ebase + n;
        #pragma unroll
        for (int d = 0; d < D_DIM; ++d)
            atomicAdd(&s_sums[s * D_DIM + d], ep[(size_t)d * N_DIM]);
    }
    __syncthreads();

    for (int i = tid; i < NSEG; i += blockDim.x)
        atomicAdd(&counts_g[b * NSEG + i], s_counts[i]);
    for (int i = tid; i < NSEG * D_DIM; i += blockDim.x)
        atomicAdd(&sums_g[b * NSEG * D_DIM + i], s_sums[i]);
}

// ---------------------------------------------------------------- kernel 2
// Second streaming pass: per-pixel hinge distance to its segment mean.
__global__ void var_pass_kernel(const float* __restrict__ emb,
                                const int* __restrict__ seg,
                                const float* __restrict__ counts_g,
                                const float* __restrict__ sums_g,
                                float* __restrict__ vsum_g) {
    __shared__ float s_mean[NSEG * D_DIM];
    __shared__ float s_vsum[NSEG];
    const int b   = blockIdx.y;
    const int tid = threadIdx.x;

    for (int i = tid; i < NSEG * D_DIM; i += blockDim.x) {
        const int l = i / D_DIM;
        s_mean[i] = sums_g[b * NSEG * D_DIM + i] / fmaxf(counts_g[b * NSEG + l], 1.0f);
    }
    for (int i = tid; i < NSEG; i += blockDim.x) s_vsum[i] = 0.0f;
    __syncthreads();

    const int pix = N_DIM / CHUNKS;
    const int n0  = blockIdx.x * pix;
    const size_t ebase = (size_t)b * D_DIM * N_DIM;

    for (int n = n0 + tid; n < n0 + pix; n += blockDim.x) {
        const float* ep = emb + ebase + n;
        __builtin_prefetch(ep + blockDim.x, 0, 0);     // global_prefetch hint
        const int s = seg[(size_t)b * N_DIM + n];
        const float* mp = &s_mean[s * D_DIM];
        float sq = 0.0f;
        #pragma unroll
        for (int d = 0; d < D_DIM; ++d) {
            const float diff = ep[(size_t)d * N_DIM] - mp[d];
            sq = fmaf(diff, diff, sq);
        }
        const float dd = (sq > 0.0f) ? sqrtf(sq) : 0.0f;  // _safe_sqrt
        float v = fmaxf(dd - DELTA_V, 0.0f);
        v = v * v;
        if (s > 0) atomicAdd(&s_vsum[s], v);
    }
    __syncthreads();

    for (int i = tid; i < NSEG; i += blockDim.x)
        atomicAdd(&vsum_g[b * NSEG + i], s_vsum[i]);
}

// ---------------------------------------------------------------- kernel 3
// One wave32 per batch: var_loss, then the 20x20 mean-pair hinge via
// V_WMMA_F32_16X16X4_F32 Gram matrix (full f32 precision).
__global__ void finalize_kernel(const float* __restrict__ counts_g,
                                const float* __restrict__ sums_g,
                                const float* __restrict__ vsum_g,
                                float* __restrict__ var_b,
                                float* __restrict__ dist_b) {
    const int b    = blockIdx.x;
    const int lane = threadIdx.x;          // 32 threads = 1 wave

    __shared__ float Mp[32][33];           // zero-padded means, conflict-free
    __shared__ float nrm[32];
    __shared__ float presLDS[32];          // presence of label i (segment i+1)

    // ---- presence / num_lanes / var_loss ----
    const float cnt  = (lane < NSEG) ? counts_g[b * NSEG + lane] : 0.0f;
    const float pres = (lane >= 1 && lane < NSEG && cnt > 0.0f) ? 1.0f : 0.0f;
    const float num_lanes = wave_reduce_add(pres);

    float vm = 0.0f;
    if (lane < NSEG) vm = vsum_g[b * NSEG + lane] / fmaxf(cnt, 1.0f) * pres;
    const float vtot = wave_reduce_add(vm);
    if (lane == 0)
        var_b[b] = (num_lanes > 0.0f) ? vtot / fmaxf(num_lanes, 1.0f) : 0.0f;

    presLDS[lane] = (lane < NLAB && counts_g[b * NSEG + lane + 1] > 0.0f) ? 1.0f : 0.0f;

    // ---- padded 32x32 mean matrix: row i = mean of label i (seg i+1) ----
    for (int i = 0; i < 32; ++i) {
        float m = 0.0f;
        if (i < NLAB)
            m = sums_g[b * NSEG * D_DIM + (i + 1) * D_DIM + lane] /
                fmaxf(counts_g[b * NSEG + i + 1], 1.0f);
        Mp[i][lane] = m;
    }
    __syncthreads();

    float nn = 0.0f;
    #pragma unroll
    for (int k = 0; k < 32; ++k) nn = fmaf(Mp[lane][k], Mp[lane][k], nn);
    nrm[lane] = nn;
    __syncthreads();

    // ---- Gram = Mp * Mp^T via 2x2 tiles of 16x16, 8 chained K=4 WMMAs ----
    const int half = lane >> 4;            // 0: lanes 0-15, 1: lanes 16-31
    const int l16  = lane & 15;
    float dl_acc = 0.0f;

    #pragma unroll
    for (int ti = 0; ti < 2; ++ti) {
        #pragma unroll
        for (int tj = 0; tj < 2; ++tj) {
            v8f c = {};
            #pragma unroll
            for (int kk = 0; kk < 8; ++kk) {
                const int kbase = kk * 4 + 2 * half;   // A VGPR0:K{0,2} VGPR1:K{1,3}
                v2f a, bb;
                a.x  = Mp[ti * 16 + l16][kbase + 0];
                a.y  = Mp[ti * 16 + l16][kbase + 1];
                bb.x = Mp[tj * 16 + l16][kbase + 0];   // B = Mp^T (rows striped)
                bb.y = Mp[tj * 16 + l16][kbase + 1];
                c = __builtin_amdgcn_wmma_f32_16x16x4_f32(
                        false, a, false, bb, (short)0, c, false, false);
            }
            // C/D layout: VGPR r -> M = r + 8*half, lane%16 -> N
            #pragma unroll
            for (int r = 0; r < 8; ++r) {
                const int i = ti * 16 + r + 8 * half;
                const int j = tj * 16 + l16;
                if (i < NLAB && j < NLAB && i != j) {
                    const float mask = presLDS[i] * presLDS[j];
                    const float sqd  = nrm[i] + nrm[j] - 2.0f * c[r];
                    const float dist = (sqd > 0.0f) ? sqrtf(sqd) : 0.0f;
                    const float h    = fmaxf(DELTA_D - dist, 0.0f);
                    dl_acc = fmaf(h * h, mask, dl_acc);
                }
            }
        }
    }

    const float dl = wave_reduce_add(dl_acc);
    if (lane == 0) {
        const float denom = fmaxf(num_lanes * (num_lanes - 1.0f), 1.0f);
        dist_b[b] = (num_lanes > 1.0f) ? dl / denom * 0.5f : 0.0f;
    }
}

// ---------------------------------------------------------------- kernel 4
__global__ void write_out_kernel(const float* __restrict__ var_b,
                                 const float* __restrict__ dist_b,
                                 float* __restrict__ out) {
    if (blockIdx.x == 0 && threadIdx.x == 0) {
        float v = 0.0f, d = 0.0f;
        for (int b = 0; b < B_DIM; ++b) { v += var_b[b]; d += dist_b[b]; }
        out[0] = v / (float)B_DIM;     // var_loss
        out[1] = d / (float)B_DIM;     // dist_loss
        out[2] = 0.0f;                 // reg_loss
    }
}

extern "C" void kernel_launch(void* const* d_in, const int* in_sizes, int n_in,
                              void* d_out, int out_size, void* d_ws, size_t ws_size,
                              hipStream_t stream) {
    const float* emb = (const float*)d_in[0];   // (8, 32, 65536) f32
    const int*   seg = (const int*)d_in[1];     // (8, 65536) i32
    float* out = (float*)d_out;                 // 3 scalars
    float* ws  = (float*)d_ws;

    float* counts = ws + OFF_COUNTS;
    float* sums   = ws + OFF_SUMS;
    float* vsum   = ws + OFF_VSUM;
    float* var_b  = ws + OFF_VAR;
    float* dist_b = ws + OFF_DIST;

    zero_ws_kernel<<<(WS_FLOATS + 255) / 256, 256, 0, stream>>>(ws, out);

    dim3 sgrid(CHUNKS, B_DIM);
    accum_kernel<<<sgrid, 256, 0, stream>>>(emb, seg, counts, sums);
    var_pass_kernel<<<sgrid, 256, 0, stream>>>(emb, seg, counts, sums, vsum);
    finalize_kernel<<<B_DIM, 32, 0, stream>>>(counts, sums, vsum, var_b, dist_b);
    write_out_kernel<<<1, 32, 0, stream>>>(var_b, dist_b, out);
}